// PartialConnection_71476845740128
// MI455X (gfx1250) — compile-verified
//
#include <hip/hip_runtime.h>
#include <stdint.h>

#define BATCH 32
#define DEG 16
#define SEG_PER_WAVE 32
#define WAVES_PER_BLOCK 4
#define THREADS (WAVES_PER_BLOCK * 32)
#define EDGES_PER_CHUNK (SEG_PER_WAVE * DEG)   // 512 edges = 2KB per array per wave

// ---------------------------------------------------------------------------
// CDNA5 async copy: global memory -> LDS, 16B per lane, tracked by ASYNCcnt.
// GV addressing mode: 64-bit address in VGPR pair, no SADDR ("off").
// ---------------------------------------------------------------------------
__device__ __forceinline__ void async_copy_b128(uint32_t lds_off, uint64_t gaddr) {
    asm volatile("global_load_async_to_lds_b128 %0, %1, off"
                 :: "v"(lds_off), "v"(gaddr)
                 : "memory");
}

__device__ __forceinline__ void wait_asynccnt0() {
    asm volatile("s_wait_asynccnt 0" ::: "memory");
}

// ---------------------------------------------------------------------------
// Transpose x (B x N, row-major) -> xT (N x B) so that the per-edge gather
// x[:, src] becomes one contiguous 128B line read across the 32 lanes.
// ---------------------------------------------------------------------------
__global__ __launch_bounds__(1024) void transpose_x_kernel(
    const float* __restrict__ x, float* __restrict__ xT, int N)
{
    __shared__ float tile[32][33];
    int nx = blockIdx.x * 32 + threadIdx.x;
    int b  = threadIdx.y;
    if (nx < N) tile[b][threadIdx.x] = x[(size_t)b * N + nx];
    __syncthreads();
    int n2 = blockIdx.x * 32 + threadIdx.y;
    if (n2 < N) xT[(size_t)n2 * BATCH + threadIdx.x] = tile[threadIdx.x][threadIdx.y];
}

// ---------------------------------------------------------------------------
// Main kernel: wave32 with lane == batch index. Each wave processes 32
// contiguous segments. Edge data (src/kernel/bias) is staged into LDS with
// async-to-LDS b128 copies (ASYNCcnt); gathers hit the L2-resident xT.
// Results go through a padded per-wave LDS tile so output stores are
// full-128B-line coalesced along n.
//
// TRANSPOSED=true : gather xsrc[idx*32 + lane]      (xsrc = xT)
// TRANSPOSED=false: gather xsrc[lane*N + idx]       (xsrc = x, workspace fallback)
// All offsets fit in uint32 (x is 12.8 MB), so address math is 32-bit.
// ---------------------------------------------------------------------------
template <bool TRANSPOSED>
__global__ __launch_bounds__(THREADS) void pc_main_kernel(
    const float* __restrict__ xsrc,
    const int*   __restrict__ edge_src,
    const float* __restrict__ kern,
    const float* __restrict__ bias,
    float* __restrict__ out,
    int N, int nChunks)
{
    __shared__ __align__(16) int   s_src[WAVES_PER_BLOCK][EDGES_PER_CHUNK];
    __shared__ __align__(16) float s_k  [WAVES_PER_BLOCK][EDGES_PER_CHUNK];
    __shared__ __align__(16) float s_b  [WAVES_PER_BLOCK][EDGES_PER_CHUNK];
    __shared__ float s_tr[WAVES_PER_BLOCK][32][33];   // stride 33: bank-conflict free

    const uint32_t tid  = threadIdx.x;
    const uint32_t wave = tid >> 5;
    const uint32_t lane = tid & 31u;
    const int chunk = blockIdx.x * WAVES_PER_BLOCK + (int)wave;
    if (chunk >= nChunks) return;

    const long e0 = (long)chunk * EDGES_PER_CHUNK;   // first edge of this wave's chunk

    // ---- Stage 6KB of edge data into this wave's LDS slice (ASYNCcnt path) ----
    {
        uint32_t ls = (uint32_t)(uintptr_t)&s_src[wave][0];
        uint32_t lk = (uint32_t)(uintptr_t)&s_k[wave][0];
        uint32_t lb = (uint32_t)(uintptr_t)&s_b[wave][0];
        uint64_t gs = (uint64_t)(uintptr_t)(edge_src + e0);
        uint64_t gk = (uint64_t)(uintptr_t)(kern + e0);
        uint64_t gb = (uint64_t)(uintptr_t)(bias + e0);
#pragma unroll
        for (int i = 0; i < 4; ++i) {
            uint32_t off = (uint32_t)(i * 512) + lane * 16u;   // 2KB per array / wave
            async_copy_b128(ls + off, gs + off);
            async_copy_b128(lk + off, gk + off);
            async_copy_b128(lb + off, gb + off);
        }
        wait_asynccnt0();   // LDS slice valid for this wave; no cross-wave sharing
    }

    // ---- Compute: lane = batch; 16-deep FMA per segment, 32-bit addressing ----
    const uint32_t laneBase = TRANSPOSED ? lane : lane * (uint32_t)N;
    for (int j = 0; j < SEG_PER_WAVE; ++j) {
        float acc = 0.0f;
#pragma unroll
        for (int d = 0; d < DEG; ++d) {
            const int e = j * DEG + d;
            const uint32_t idx = (uint32_t)s_src[wave][e];      // wave-uniform (LDS broadcast)
            const uint32_t off = TRANSPOSED ? (idx * (uint32_t)BATCH + laneBase)
                                            : (laneBase + idx);
            const float xv = xsrc[off];
            acc = fmaf(xv, s_k[wave][e], acc + s_b[wave][e]);
        }
        s_tr[wave][j][lane] = acc;                              // [segment][batch]
    }

    // ---- Transposed store: lanes sweep n so each store is one 128B line ----
    const uint32_t obase = (uint32_t)(chunk * SEG_PER_WAVE) + lane;
#pragma unroll 4
    for (uint32_t b = 0; b < BATCH; ++b) {
        out[b * (uint32_t)N + obase] = s_tr[wave][lane][b];
    }
}

// Scalar tail for N not divisible by 32 (not hit for N = 100000, kept for safety).
__global__ __launch_bounds__(32) void pc_tail_kernel(
    const float* __restrict__ xsrc, long sIdx, long sLane,
    const int* __restrict__ edge_src,
    const float* __restrict__ kern,
    const float* __restrict__ bias,
    float* __restrict__ out, int N, int segStart)
{
    const int n = segStart + blockIdx.x;
    const int lane = threadIdx.x;
    if (n >= N) return;
    float acc = 0.0f;
#pragma unroll
    for (int d = 0; d < DEG; ++d) {
        const long e = (long)n * DEG + d;
        const int idx = edge_src[e];
        acc = fmaf(xsrc[(long)idx * sIdx + (long)lane * sLane], kern[e], acc + bias[e]);
    }
    out[(size_t)lane * N + n] = acc;
}

extern "C" void kernel_launch(void* const* d_in, const int* in_sizes, int n_in,
                              void* d_out, int out_size, void* d_ws, size_t ws_size,
                              hipStream_t stream) {
    (void)n_in; (void)out_size;
    const float* x        = (const float*)d_in[0];
    const float* kern     = (const float*)d_in[1];
    const float* bias     = (const float*)d_in[2];
    const int*   edge_src = (const int*)d_in[3];
    // d_in[4] = segment_ids (implicit: e / DEG, unused), d_in[5] = num_units scalar
    float* out = (float*)d_out;

    const int N = in_sizes[0] / BATCH;   // x is (B, N) flat

    // Prefer the transposed-x gather path when workspace permits.
    const size_t xtBytes = (size_t)N * BATCH * sizeof(float);
    const bool useT = (ws_size >= xtBytes);

    const float* xsrc = x;
    if (useT) {
        float* xT = (float*)d_ws;
        dim3 tb(32, 32);
        int tg = (N + 31) / 32;
        transpose_x_kernel<<<tg, tb, 0, stream>>>(x, xT, N);
        xsrc = xT;
    }

    const int nChunks = N / SEG_PER_WAVE;
    if (nChunks > 0) {
        const int blocks = (nChunks + WAVES_PER_BLOCK - 1) / WAVES_PER_BLOCK;
        if (useT) {
            pc_main_kernel<true><<<blocks, THREADS, 0, stream>>>(
                xsrc, edge_src, kern, bias, out, N, nChunks);
        } else {
            pc_main_kernel<false><<<blocks, THREADS, 0, stream>>>(
                xsrc, edge_src, kern, bias, out, N, nChunks);
        }
    }
    const int rem = N - nChunks * SEG_PER_WAVE;
    if (rem > 0) {
        const long sIdx  = useT ? BATCH : 1;
        const long sLane = useT ? 1 : N;
        pc_tail_kernel<<<rem, 32, 0, stream>>>(
            xsrc, sIdx, sLane, edge_src, kern, bias, out, N, nChunks * SEG_PER_WAVE);
    }
}